// BayesianFilter_8718783611687
// MI455X (gfx1250) — compile-verified
//
#include <hip/hip_runtime.h>
#include <math.h>

typedef __attribute__((ext_vector_type(2))) float v2f;
typedef __attribute__((ext_vector_type(4))) float v4f;
typedef __attribute__((ext_vector_type(8))) float v8f;

#define NBLOCKS   256
#define WAVES_PB  8
#define ITERS     2          // 256*8*2 wave-tasks * 16 samples = 65536
#define ACC_N     17         // 16 weighted-noise sums + 1 score sum

// Bernstein basis value: C(n,k) * s^k * (1-s)^(n-k), s = p/255
__device__ __forceinline__ float bern(int p, int k, int n) {
    if (k > n) return 0.0f;
    float s = (float)p * (1.0f / 255.0f);
    float comb = 1.0f;
    for (int i = 1; i <= k; ++i) comb = comb * (float)(n - i + 1) / (float)i;
    float r = comb;
    for (int i = 0; i < k; ++i) r *= s;
    for (int i = 0; i < n - k; ++i) r *= (1.0f - s);
    return r;
}

__global__ __launch_bounds__(256) void bf_main(
    const float* __restrict__ curve,     // [8,2] = 16
    const float* __restrict__ noise,     // [65536,8,2]
    const float* __restrict__ sp_tab,    // [16] speeds
    const float* __restrict__ br_tab,    // [16] braking limits
    const float* __restrict__ deltaT,    // [1]
    float* __restrict__ part)            // [NBLOCKS, ACC_N] partials
{
    // A-fragment tables in ISA 16x4 f32 A-layout:
    // entry e = (tile*2 + chunk)*32 + lane ; holds {bern(p,kb), bern(p,kb+1)}
    // p = 16*tile + lane%16 ; kb = 4*chunk + 2*(lane/16)
    __shared__ float ldsA1[16 * 2 * 32 * 2];   // order-6 basis (M_D1), K padded to 8
    __shared__ float ldsA2[16 * 2 * 32 * 2];   // order-5 basis (M_D2), K padded to 8
    __shared__ v2f   ldsFp2[15];               // {fp[i], fp[i+1]-fp[i]} per segment
    __shared__ float wavePart[WAVES_PB][ACC_N];

    const int tid = threadIdx.x;
    for (int e = tid; e < 1024; e += 256) {
        int t = e >> 6, c = (e >> 5) & 1, l = e & 31;
        int p  = t * 16 + (l & 15);
        int kb = c * 4 + ((l >> 4) << 1);
        ldsA1[e * 2 + 0] = bern(p, kb,     6);
        ldsA1[e * 2 + 1] = bern(p, kb + 1, 6);
        ldsA2[e * 2 + 0] = bern(p, kb,     5);
        ldsA2[e * 2 + 1] = bern(p, kb + 1, 5);
    }
    if (tid < 15) {
        float f0 = br_tab[tid], f1 = br_tab[tid + 1];
        ldsFp2[tid] = (v2f){f0, f1 - f0};
    }
    __syncthreads();

    const int lane = tid & 31;
    const int wave = tid >> 5;
    const int n    = lane & 15;          // D/B column == local sample 0..15
    const int h    = lane >> 4;          // lane-half selects K-high / M-high

    const float dt   = deltaT[0];
    const float idt  = 1.0f / dt;
    const float idt2 = idt * idt;
    const float x0 = sp_tab[0], x1 = sp_tab[1], xlast = sp_tab[15];
    const float invh = 1.0f / (x1 - x0);

    float cvx[8], cvy[8];
#pragma unroll
    for (int j = 0; j < 8; ++j) { cvx[j] = curve[j * 2]; cvy[j] = curve[j * 2 + 1]; }

    float accx[8], accy[8];
#pragma unroll
    for (int j = 0; j < 8; ++j) { accx[j] = 0.0f; accy[j] = 0.0f; }
    float spAcc = 0.0f;

    for (int it = 0; it < ITERS; ++it) {
        const int gw     = it * (NBLOCKS * WAVES_PB) + blockIdx.x * WAVES_PB + wave;
        const int sample = gw * 16 + n;

        // one 64B noise row per lane: 4 coalesced b128 loads
        const v4f* rowp = (const v4f*)(noise + (size_t)sample * 16);
        v4f r0 = rowp[0], r1 = rowp[1], r2 = rowp[2], r3 = rowp[3];
        float nzx[8], nzy[8];
        nzx[0] = r0.x; nzy[0] = r0.y; nzx[1] = r0.z; nzy[1] = r0.w;
        nzx[2] = r1.x; nzy[2] = r1.y; nzx[3] = r1.z; nzy[3] = r1.w;
        nzx[4] = r2.x; nzy[4] = r2.y; nzx[5] = r2.z; nzy[5] = r2.w;
        nzx[6] = r3.x; nzy[6] = r3.y; nzx[7] = r3.z; nzy[7] = r3.w;

        float cux[8], cuy[8];
#pragma unroll
        for (int j = 0; j < 8; ++j) { cux[j] = cvx[j] + nzx[j]; cuy[j] = cvy[j] + nzy[j]; }

        // derivative control points, scaled by 1/dT and 1/dT^2, K padded with 0
        float dPx[8], dPy[8], d2Px[8], d2Py[8];
#pragma unroll
        for (int j = 0; j < 7; ++j) {
            dPx[j] = 7.0f * (cux[j + 1] - cux[j]) * idt;
            dPy[j] = 7.0f * (cuy[j + 1] - cuy[j]) * idt;
        }
        dPx[7] = 0.0f; dPy[7] = 0.0f;
#pragma unroll
        for (int j = 0; j < 6; ++j) {
            d2Px[j] = 42.0f * (cux[j + 2] - 2.0f * cux[j + 1] + cux[j]) * idt2;
            d2Py[j] = 42.0f * (cuy[j + 2] - 2.0f * cuy[j + 1] + cuy[j]) * idt2;
        }
        d2Px[6] = 0.0f; d2Px[7] = 0.0f; d2Py[6] = 0.0f; d2Py[7] = 0.0f;

        // B fragments, 4x16 f32 layout: column = lane%16, K = vgpr + 2*(lane/16)
        v2f bvx0 = h ? (v2f){dPx[2],  dPx[3]}  : (v2f){dPx[0],  dPx[1]};
        v2f bvx1 = h ? (v2f){dPx[6],  dPx[7]}  : (v2f){dPx[4],  dPx[5]};
        v2f bvy0 = h ? (v2f){dPy[2],  dPy[3]}  : (v2f){dPy[0],  dPy[1]};
        v2f bvy1 = h ? (v2f){dPy[6],  dPy[7]}  : (v2f){dPy[4],  dPy[5]};
        v2f bax0 = h ? (v2f){d2Px[2], d2Px[3]} : (v2f){d2Px[0], d2Px[1]};
        v2f bax1 = h ? (v2f){d2Px[6], d2Px[7]} : (v2f){d2Px[4], d2Px[5]};
        v2f bay0 = h ? (v2f){d2Py[2], d2Py[3]} : (v2f){d2Py[0], d2Py[1]};
        v2f bay1 = h ? (v2f){d2Py[6], d2Py[7]} : (v2f){d2Py[4], d2Py[5]};

        float violSum = 0.0f;
        float camax2  = 0.0f;

        for (int t = 0; t < 16; ++t) {
            const int base = (t * 2) * 32 + lane;
            v2f a10 = *(const v2f*)&ldsA1[base * 2];
            v2f a11 = *(const v2f*)&ldsA1[(base + 32) * 2];
            v2f a20 = *(const v2f*)&ldsA2[base * 2];
            v2f a21 = *(const v2f*)&ldsA2[(base + 32) * 2];

            v8f vtx = {}, vty = {}, atx = {}, aty = {};
            vtx = __builtin_amdgcn_wmma_f32_16x16x4_f32(false, a10, false, bvx0,
                                                        (short)0, vtx, false, false);
            vtx = __builtin_amdgcn_wmma_f32_16x16x4_f32(false, a11, false, bvx1,
                                                        (short)0, vtx, false, false);
            vty = __builtin_amdgcn_wmma_f32_16x16x4_f32(false, a10, false, bvy0,
                                                        (short)0, vty, false, false);
            vty = __builtin_amdgcn_wmma_f32_16x16x4_f32(false, a11, false, bvy1,
                                                        (short)0, vty, false, false);
            atx = __builtin_amdgcn_wmma_f32_16x16x4_f32(false, a20, false, bax0,
                                                        (short)0, atx, false, false);
            atx = __builtin_amdgcn_wmma_f32_16x16x4_f32(false, a21, false, bax1,
                                                        (short)0, atx, false, false);
            aty = __builtin_amdgcn_wmma_f32_16x16x4_f32(false, a20, false, bay0,
                                                        (short)0, aty, false, false);
            aty = __builtin_amdgcn_wmma_f32_16x16x4_f32(false, a21, false, bay1,
                                                        (short)0, aty, false, false);

#pragma unroll
            for (int r = 0; r < 8; ++r) {
                float vx = vtx[r], vy = vty[r];
                float ax = atx[r], ay = aty[r];
                float speed = sqrtf(vx * vx + vy * vy);
                float invsp = 1.0f / speed;
                float lin   = (ax * vx + ay * vy) * invsp;
                // piecewise-linear braking limit (uniform table spacing)
                float xc = fminf(fmaxf(speed, x0), xlast);
                float tp = (xc - x0) * invh;
                int   i  = (int)tp;  i = min(i, 14);
                v2f   ab = ldsFp2[i];
                float lim = ab.x + (tp - (float)i) * ab.y;
                float viol = fmaxf(fminf(lin - lim, 0.0f), -1e9f);
                violSum += viol;
                float cx = ax - vx * invsp * lin;
                float cy = ay - vy * invsp * lin;
                camax2 = fmaxf(camax2, cx * cx + cy * cy);
            }
        }
        // combine the two row-halves (M 0..7 in lanes<16, M 8..15 in lanes>=16)
        violSum += __shfl_xor(violSum, 16, 32);
        camax2   = fmaxf(camax2, __shfl_xor(camax2, 16, 32));

        float brake = fminf(fmaxf(expf(violSum * (1.0f / 256.0f)), 0.0f), 1.0f);
        float camax = sqrtf(camax2);
        float cas   = fminf(fmaxf(expf(fmaxf(camax - 19.6f, 0.0f)), 0.0f), 1.0f);
        float sp    = brake * cas;

#pragma unroll
        for (int j = 0; j < 8; ++j) {
            accx[j] += sp * nzx[j];
            accy[j] += sp * nzy[j];
        }
        spAcc += sp;   // lanes 16..31 duplicate; reduction below only sums 16 lanes
    }

    // deterministic in-wave reduction over the 16 samples (fixed shuffle order);
    // lane L accumulates lanes {L^1, L^2, L^4, L^8} -> lane 0 = sum of lanes 0..15
#pragma unroll
    for (int m = 1; m <= 8; m <<= 1) {
#pragma unroll
        for (int j = 0; j < 8; ++j) {
            accx[j] += __shfl_xor(accx[j], m, 32);
            accy[j] += __shfl_xor(accy[j], m, 32);
        }
        spAcc += __shfl_xor(spAcc, m, 32);
    }
    if (lane == 0) {
#pragma unroll
        for (int j = 0; j < 8; ++j) {
            wavePart[wave][j * 2 + 0] = accx[j];
            wavePart[wave][j * 2 + 1] = accy[j];
        }
        wavePart[wave][16] = spAcc;
    }
    __syncthreads();

    // fixed-order block reduction; every slot rewritten each launch
    if (tid < ACC_N) {
        float s = 0.0f;
#pragma unroll
        for (int w = 0; w < WAVES_PB; ++w) s += wavePart[w][tid];
        part[blockIdx.x * ACC_N + tid] = s;
    }
}

__global__ void bf_final(const float* __restrict__ curve,
                         const float* __restrict__ part,
                         float* __restrict__ out)
{
    __shared__ float tot[ACC_N];
    const int tid = threadIdx.x;
    if (tid < ACC_N) {
        float s = 0.0f;
        for (int b = 0; b < NBLOCKS; ++b) s += part[b * ACC_N + tid];
        tot[tid] = s;
    }
    __syncthreads();
    if (tid < 16) {
        // sum_s probs*curves = curve + (sum sp*noise)/(sum sp)
        out[tid] = curve[tid] + tot[tid] / tot[16];
    }
}

extern "C" void kernel_launch(void* const* d_in, const int* in_sizes, int n_in,
                              void* d_out, int out_size, void* d_ws, size_t ws_size,
                              hipStream_t stream) {
    const float* curve = (const float*)d_in[0];
    const float* noise = (const float*)d_in[1];
    const float* spt   = (const float*)d_in[2];
    const float* brt   = (const float*)d_in[3];
    const float* dT    = (const float*)d_in[4];
    float* part = (float*)d_ws;            // NBLOCKS * ACC_N floats (~17 KB)
    float* out  = (float*)d_out;           // [8,2] f32

    bf_main <<<NBLOCKS, 256, 0, stream>>>(curve, noise, spt, brt, dT, part);
    bf_final<<<1, 32, 0, stream>>>(curve, part, out);
}